// LaneATT_89180700934362
// MI455X (gfx1250) — compile-verified
//
#include <hip/hip_runtime.h>

// ---------------------------------------------------------------------------
// LaneATT attention on MI455X (gfx1250, wave32).
// fp32 WMMA (V_WMMA_F32_16X16X4_F32), exact fp32 semantics vs the reference.
// Per-batch pipeline: gather -> gemm1(+bias, shifted scatter) -> softmax -> gemm2.
//  * Operands zero-padded once per launch -> no predication in GEMM loops.
//  * B operands pair-packed (rows 2p,2p+1 interleaved by column) so each
//    B-frag is ONE contiguous global_load_b64.
//  * A-tiles staged in LDS, shared by the 4 waves of a block (ds_load b64).
//  * 4 N-tiles per wave: 8 WMMA per ds_load_2addr_b64 + 8 b64 loads.
// ---------------------------------------------------------------------------

typedef __attribute__((ext_vector_type(2))) float v2f;
typedef __attribute__((ext_vector_type(8))) float v8f;

#define B_DIM  32
#define A_DIM  1000
#define C_DIM  64
#define H_DIM  12
#define W_DIM  20
#define CH     768          // C_DIM * H_DIM
#define N1     999          // A_DIM - 1 (valid score columns)
#define MP     1008         // 63*16 padded row count
#define NP1    1024         // padded score-column count (Wpad/bias/attn cols)
#define FP     1024         // padded feats row count (gemm2 K)
#define LDATT  1024         // attn leading dimension
#define KC     64           // K-chunk staged in LDS
#define LDS_PAD 68          // LDS row stride (floats): 64+4 -> conflict-free

// Pair-packed index: element (k, n) of a K x N matrix lives at
//   Pk[(k>>1) * 2N + 2n + (k&1)]
// so rows (2p, 2p+1) at column n are adjacent -> v2f B-frag load.

// ---------------------------------------------------------------------------
// One-time: W_att (CH x N1) -> pair-packed Wpk (CH x NP1), zero cols >= N1.
// ---------------------------------------------------------------------------
__global__ void pad_w(const float* __restrict__ Wt, float* __restrict__ Wpk)
{
    int e = blockIdx.x * blockDim.x + threadIdx.x;
    if (e >= CH * NP1) return;
    int k = e >> 10;            // / NP1
    int j = e & (NP1 - 1);
    float v = (j < N1) ? Wt[k * N1 + j] : 0.0f;
    Wpk[(size_t)(k >> 1) * (2 * NP1) + 2 * j + (k & 1)] = v;
}

__global__ void pad_bias(const float* __restrict__ bias, float* __restrict__ biasp)
{
    int j = blockIdx.x * blockDim.x + threadIdx.x;
    if (j < NP1) biasp[j] = (j < N1) ? bias[j] : 0.0f;
}

// ---------------------------------------------------------------------------
// Gather feats[a][k] = rois[a, c, h] (k = c*H + h), zero-masked, zero rows
// a >= A_DIM up to FP.  Writes row-major copy (gemm1 A) and pair-packed copy
// (gemm2 B) in one pass.
// ---------------------------------------------------------------------------
__global__ void gather_feats(const float* __restrict__ fv,   // (B, C, H, W)
                             const int*   __restrict__ zi,   // (A, C, H)
                             const int*   __restrict__ yi,
                             const int*   __restrict__ xi,
                             const int*   __restrict__ mask,
                             float* __restrict__ feats,      // (FP, CH) row-major
                             float* __restrict__ featsPk,    // (FP, CH) pair-packed
                             int b)
{
    int e = blockIdx.x * blockDim.x + threadIdx.x;
    if (e >= FP * CH) return;
    int a = e / CH;
    int k = e - a * CH;                 // channel column, == c*H + h
    float v = 0.0f;
    if (a < A_DIM) {
        int idx = a * CH + k;           // (a*C + c)*H + h
        if (!mask[idx]) {
            int z = zi[idx], y = yi[idx], x = xi[idx];
            v = fv[((b * C_DIM + z) * H_DIM + y) * W_DIM + x];
        }
    }
    feats[e] = v;
    featsPk[(size_t)(a >> 1) * (2 * CH) + 2 * k + (a & 1)] = v;
}

// ---------------------------------------------------------------------------
// Cooperative 16 x KC A-tile stage: 128 threads, two float4 loads each.
// ---------------------------------------------------------------------------
__device__ __forceinline__ void stage_a(float (*As)[LDS_PAD],
                                        const float* __restrict__ Abase,
                                        int i0, int lda, int kc, int tid)
{
    const int r = tid >> 3;             // 0..15 row
    const int g = (tid & 7) * 8;        // 0,8,..,56 col group
    const float* src = Abase + (size_t)(i0 + r) * lda + kc + g;
    float4 p0 = *(const float4*)(src);
    float4 p1 = *(const float4*)(src + 4);
    *(float4*)&As[r][g]     = p0;
    *(float4*)&As[r][g + 4] = p1;
}

// ---------------------------------------------------------------------------
// GEMM1: S = feats(MP x CH) @ W(CH x NP1) + bias, scattered into
// attn[i][jp + (jp >= i)].  Each wave owns a 16 x 64 output slab (4 tiles).
// fp32 WMMA lane layout (ISA 7.12.2), half = lane>>4, n = lane&15:
//   A: a[j]   = A[lane&15][2*half + j]
//   B: b[j]   = B[2*half + j][n]        (one v2f load from pair-packed W)
//   C: acc[r] = C[r + 8*half][n]
// ---------------------------------------------------------------------------
__global__ void gemm1_scores(const float* __restrict__ feats, // (MP.., CH)
                             const float* __restrict__ Wpk,   // (CH, NP1) packed
                             const float* __restrict__ biasp, // (NP1)
                             float* __restrict__ attn)        // (MP, LDATT)
{
    __shared__ float As[16][LDS_PAD];

    const int lane = threadIdx.x;                  // 0..31
    const int wave = threadIdx.y;                  // 0..3
    const int tid  = wave * 32 + lane;
    const int q    = blockIdx.x * 4 + wave;        // N-quad 0..15
    const int j0   = q * 64;                       // first of 4 col tiles
    const int i0   = blockIdx.y * 16;              // row tile base
    const int half = lane >> 4;
    const int n    = lane & 15;
    const int ar   = lane & 15;                    // A-frag row

    v8f acc0, acc1, acc2, acc3;
    {
        const float b0 = biasp[j0 + n];
        const float b1 = biasp[j0 + 16 + n];
        const float b2 = biasp[j0 + 32 + n];
        const float b3 = biasp[j0 + 48 + n];
#pragma unroll
        for (int r = 0; r < 8; ++r) {
            acc0[r] = b0; acc1[r] = b1; acc2[r] = b2; acc3[r] = b3;
        }
    }

    for (int kc = 0; kc < CH; kc += KC) {
        __syncthreads();
        stage_a(As, feats, i0, CH, kc, tid);
        __syncthreads();
#pragma unroll
        for (int kk = 0; kk < KC; kk += 4) {
            const int ka = kk + 2 * half;              // even
            v2f afrag = *(const v2f*)&As[ar][ka];
            const float* wrow = Wpk + (size_t)((kc + ka) >> 1) * (2 * NP1);
            v2f b0 = *(const v2f*)&wrow[2 * (j0 + n)];
            v2f b1 = *(const v2f*)&wrow[2 * (j0 + 16 + n)];
            v2f b2 = *(const v2f*)&wrow[2 * (j0 + 32 + n)];
            v2f b3 = *(const v2f*)&wrow[2 * (j0 + 48 + n)];
            acc0 = __builtin_amdgcn_wmma_f32_16x16x4_f32(
                false, afrag, false, b0, (short)0, acc0, false, false);
            acc1 = __builtin_amdgcn_wmma_f32_16x16x4_f32(
                false, afrag, false, b1, (short)0, acc1, false, false);
            acc2 = __builtin_amdgcn_wmma_f32_16x16x4_f32(
                false, afrag, false, b2, (short)0, acc2, false, false);
            acc3 = __builtin_amdgcn_wmma_f32_16x16x4_f32(
                false, afrag, false, b3, (short)0, acc3, false, false);
        }
    }

    // Epilogue: scatter with the diagonal-skip column shift.
#pragma unroll
    for (int t = 0; t < 4; ++t) {
        const int jp = j0 + t * 16 + n;
        if (jp < N1) {
#pragma unroll
            for (int r = 0; r < 8; ++r) {
                const int m = i0 + r + 8 * half;       // global row i
                const int jcol = jp + (jp >= m ? 1 : 0);
                const float val = (t == 0) ? acc0[r] : (t == 1) ? acc1[r]
                                : (t == 2) ? acc2[r] : acc3[r];
                attn[(size_t)m * LDATT + jcol] = val;
            }
        }
    }
}

// ---------------------------------------------------------------------------
// Row softmax over the 999 off-diagonal logits; zero diagonal + padding cols
// (also scrubs stale data from the previous batch).  256 threads per row.
// ---------------------------------------------------------------------------
__global__ void softmax_row(float* __restrict__ attn)
{
    __shared__ float red[256];
    const int i = blockIdx.x;            // row, 0..A_DIM-1
    const int t = threadIdx.x;
    float* row = attn + (size_t)i * LDATT;

    float mx = -3.402823466e+38f;
    for (int j = t; j < A_DIM; j += 256)
        if (j != i) mx = fmaxf(mx, row[j]);
    red[t] = mx;
    __syncthreads();
    for (int s = 128; s > 0; s >>= 1) {
        if (t < s) red[t] = fmaxf(red[t], red[t + s]);
        __syncthreads();
    }
    mx = red[0];
    __syncthreads();

    float sum = 0.0f;
    for (int j = t; j < A_DIM; j += 256)
        if (j != i) sum += expf(row[j] - mx);
    red[t] = sum;
    __syncthreads();
    for (int s = 128; s > 0; s >>= 1) {
        if (t < s) red[t] += red[t + s];
        __syncthreads();
    }
    const float inv = 1.0f / red[0];

    for (int j = t; j < A_DIM; j += 256)
        row[j] = (j == i) ? 0.0f : expf(row[j] - mx) * inv;
    for (int j = A_DIM + t; j < LDATT; j += 256)    // zero diag-pad columns
        row[j] = 0.0f;
}

// ---------------------------------------------------------------------------
// GEMM2: out = attn(MP x FP) @ feats(FP x CH); K padded to 1024 (zeros).
// B from pair-packed feats; masked stores for rows >= A_DIM.
// ---------------------------------------------------------------------------
__global__ void gemm2_out(const float* __restrict__ attn,     // (MP, LDATT)
                          const float* __restrict__ featsPk,  // (FP, CH) packed
                          float* __restrict__ outb)           // (A_DIM, CH)
{
    __shared__ float As[16][LDS_PAD];

    const int lane = threadIdx.x;
    const int wave = threadIdx.y;
    const int tid  = wave * 32 + lane;
    const int q    = blockIdx.x * 4 + wave;        // N-quad 0..11
    const int n0   = q * 64;
    const int i0   = blockIdx.y * 16;
    const int half = lane >> 4;
    const int n    = lane & 15;
    const int ar   = lane & 15;

    v8f acc0 = {}, acc1 = {}, acc2 = {}, acc3 = {};

    for (int kc = 0; kc < FP; kc += KC) {
        __syncthreads();
        stage_a(As, attn, i0, LDATT, kc, tid);
        __syncthreads();
#pragma unroll
        for (int kk = 0; kk < KC; kk += 4) {
            const int ka = kk + 2 * half;
            v2f afrag = *(const v2f*)&As[ar][ka];
            const float* frow = featsPk + (size_t)((kc + ka) >> 1) * (2 * CH);
            v2f b0 = *(const v2f*)&frow[2 * (n0 + n)];
            v2f b1 = *(const v2f*)&frow[2 * (n0 + 16 + n)];
            v2f b2 = *(const v2f*)&frow[2 * (n0 + 32 + n)];
            v2f b3 = *(const v2f*)&frow[2 * (n0 + 48 + n)];
            acc0 = __builtin_amdgcn_wmma_f32_16x16x4_f32(
                false, afrag, false, b0, (short)0, acc0, false, false);
            acc1 = __builtin_amdgcn_wmma_f32_16x16x4_f32(
                false, afrag, false, b1, (short)0, acc1, false, false);
            acc2 = __builtin_amdgcn_wmma_f32_16x16x4_f32(
                false, afrag, false, b2, (short)0, acc2, false, false);
            acc3 = __builtin_amdgcn_wmma_f32_16x16x4_f32(
                false, afrag, false, b3, (short)0, acc3, false, false);
        }
    }

#pragma unroll
    for (int r = 0; r < 8; ++r) {
        const int m = i0 + r + 8 * half;
        if (m < A_DIM) {
            float* orow = outb + (size_t)m * CH;
            orow[n0 + n]      = acc0[r];
            orow[n0 + 16 + n] = acc1[r];
            orow[n0 + 32 + n] = acc2[r];
            orow[n0 + 48 + n] = acc3[r];
        }
    }
}

// ---------------------------------------------------------------------------
// Launch.  Workspace layout (floats):
//   Wpk   : CH*NP1   (3.1 MB)    biasp  : NP1
//   feats : FP*CH    (3.1 MB)    featsPk: FP*CH (3.1 MB)
//   attn  : MP*LDATT (4.1 MB)                          total ~13.5 MB
// ---------------------------------------------------------------------------
extern "C" void kernel_launch(void* const* d_in, const int* in_sizes, int n_in,
                              void* d_out, int out_size, void* d_ws, size_t ws_size,
                              hipStream_t stream)
{
    const float* fv   = (const float*)d_in[0];   // feature_volumes (B,C,H,W) f32
    const int*   zi   = (const int*)  d_in[1];
    const int*   yi   = (const int*)  d_in[2];
    const int*   xi   = (const int*)  d_in[3];
    const int*   mask = (const int*)  d_in[4];
    const float* Wt   = (const float*)d_in[5];   // (CH, N1) f32
    const float* bias = (const float*)d_in[6];   // (N1) f32
    float*       out  = (float*)d_out;           // (B, A, CH) f32

    float* Wpk     = (float*)d_ws;
    float* biasp   = Wpk     + (size_t)CH * NP1;
    float* feats   = biasp   + NP1;
    float* featsPk = feats   + (size_t)FP * CH;
    float* attn    = featsPk + (size_t)FP * CH;

    const dim3 wblk(32, 4);                      // 4 wave32s per block

    // One-time operand padding/packing (batch-invariant).
    pad_w<<<(CH * NP1 + 255) / 256, 256, 0, stream>>>(Wt, Wpk);
    pad_bias<<<(NP1 + 255) / 256, 256, 0, stream>>>(bias, biasp);

    for (int b = 0; b < B_DIM; ++b) {
        gather_feats<<<(FP * CH + 255) / 256, 256, 0, stream>>>(
            fv, zi, yi, xi, mask, feats, featsPk, b);
        gemm1_scores<<<dim3(4, 63), wblk, 0, stream>>>(feats, Wpk, biasp, attn);
        softmax_row<<<dim3(A_DIM), 256, 0, stream>>>(attn);
        gemm2_out<<<dim3(3, 63), wblk, 0, stream>>>(attn, featsPk,
                                                    out + (size_t)b * A_DIM * CH);
    }
}